// GHyperLayer_58763742544527
// MI455X (gfx1250) — compile-verified
//
#include <hip/hip_runtime.h>
#include <stdint.h>

#define BDIM 32
#define KDIM 256
#define PPK 36
#define PDIM (KDIM * PPK)      // 9216 points per batch
#define MT_COUNT (PDIM / 16)   // 576 M-tiles per batch
#define OUTSZ 1024
#define INSZ 1024
#define HSIZE 16384
#define HMASK (HSIZE - 1)
#define EPSF 1e-6f
// -0.5 * log2(e): folded into the B-vectors so props = exp2(dot + const)
#define NEG_HALF_LOG2E (-0.72134752044448170368f)

typedef __attribute__((ext_vector_type(2))) float v2f;
typedef __attribute__((ext_vector_type(8))) float v8f;

// ---------------------------------------------------------------- init
__global__ void k_init(unsigned long long* __restrict__ hkey,
                       int* __restrict__ hidx,
                       float* __restrict__ colsum,
                       float* __restrict__ y) {
  int i = blockIdx.x * blockDim.x + threadIdx.x;
  if (i < BDIM * HSIZE) { hkey[i] = 0ull; hidx[i] = 0x7FFFFFFF; }
  if (i < BDIM * OUTSZ) y[i] = 0.0f;
  if (i < BDIM * KDIM)  colsum[i] = 0.0f;
}

// ---------------------------------------------------------------- prep
// Per (b,k): cvec[8] = S*{c0, -2c0m0, c1, -2c1m1, const}, value, w, 0
//            with S = -0.5*log2(e), so density = exp2(A.B + const)
// Per point: feat[4] = {p0^2, p0, p1^2, p1}, key = (p0+1)^2 (p1+1)^3
__global__ void k_prep(const float* __restrict__ res,
                       const float* __restrict__ rr_u,
                       const float* __restrict__ g_u,
                       float* __restrict__ feat,
                       unsigned long long* __restrict__ keys,
                       float* __restrict__ cvec) {
  int t = blockIdx.x * blockDim.x + threadIdx.x;
  if (t >= BDIM * KDIM) return;
  int b = t / KDIM, k = t % KDIM;

  const float* r = res + (size_t)t * 4;
  float m0 = 1023.0f / (1.0f + __expf(-r[0]));   // sigmoid * (RNG-1)
  float m1 = 1023.0f / (1.0f + __expf(-r[1]));
  float si = r[2] + 2.0f;                          // + SIGMA_BOOST
  float sp = fmaxf(si, 0.0f) + log1pf(__expf(-fabsf(si)));  // softplus
  float sig = (sp + EPSF) * 1024.0f;               // same both dims (RNG equal)
  float c0 = 1.0f / (EPSF + sig);
  float c1 = c0;

  const float S = NEG_HALF_LOG2E;
  float* cv = cvec + (size_t)t * 8;
  cv[0] = S * c0;  cv[1] = S * -2.0f * c0 * m0;
  cv[2] = S * c1;  cv[3] = S * -2.0f * c1 * m1;
  cv[4] = S * (c0 * m0 * m0 + c1 * m1 * m1);
  cv[5] = r[3];    cv[6] = 0.0f;  cv[7] = 0.0f;

  int base = b * PDIM + k * PPK;
  auto emit = [&](int j, int p0, int p1) {
    int row = base + j;
    float* f = feat + (size_t)row * 4;
    float a0 = (float)p0, a1 = (float)p1;
    f[0] = a0 * a0; f[1] = a0; f[2] = a1 * a1; f[3] = a1;
    unsigned long long u0 = (unsigned long long)(p0 + 1);
    unsigned long long u1 = (unsigned long long)(p1 + 1);
    keys[row] = u0 * u0 * u1 * u1 * u1;
  };

  // 4 floor/ceil neighbors, FLOOR_MASK order: TT, TF, FT, FF
  int f0 = (int)floorf(m0), f1 = (int)floorf(m1);
  int e0 = (int)ceilf(m0),  e1 = (int)ceilf(m1);
  emit(0, f0, f1); emit(1, f0, e1); emit(2, e0, f1); emit(3, e0, e1);

  const float ONE = 1.0f - EPSF;
  // 16 global uniform samples
  const float* gp = g_u + (size_t)t * 32;
  for (int j = 0; j < 16; ++j) {
    int p0 = (int)floorf(gp[j * 2 + 0] * ONE * 1024.0f);
    int p1 = (int)floorf(gp[j * 2 + 1] * ONE * 1024.0f);
    emit(4 + j, p0, p1);
  }
  // 16 region samples around round(mean)
  float mn0 = rintf(m0), mn1 = rintf(m1);
  float lo0 = mn0 - 8.0f; if (lo0 < 0.0f) lo0 = 0.0f; if (mn0 + 8.0f > 1024.0f) lo0 = 1008.0f;
  float lo1 = mn1 - 8.0f; if (lo1 < 0.0f) lo1 = 0.0f; if (mn1 + 8.0f > 1024.0f) lo1 = 1008.0f;
  const float* rp = rr_u + (size_t)t * 32;
  for (int j = 0; j < 16; ++j) {
    int p0 = (int)(rp[j * 2 + 0] * ONE * 16.0f + lo0);
    int p1 = (int)(rp[j * 2 + 1] * ONE * 16.0f + lo1);
    emit(20 + j, p0, p1);
  }
}

// ---------------------------------------------------------------- dup detect
__device__ __forceinline__ unsigned int hash_slot(unsigned long long key) {
  return (unsigned int)((key * 0x9E3779B97F4A7C15ull) >> 50) & HMASK;
}

__global__ void k_insert(const unsigned long long* __restrict__ keys,
                         unsigned long long* __restrict__ hkey,
                         int* __restrict__ hidx) {
  int i = blockIdx.x * blockDim.x + threadIdx.x;
  if (i >= BDIM * PDIM) return;
  int b = i / PDIM, idx = i % PDIM;
  unsigned long long key = keys[i];
  unsigned long long* hk = hkey + (size_t)b * HSIZE;
  int* hi = hidx + (size_t)b * HSIZE;
  unsigned int s = hash_slot(key);
  for (int probe = 0; probe < HSIZE; ++probe) {
    unsigned long long old = atomicCAS(&hk[s], 0ull, key);
    if (old == 0ull || old == key) { atomicMin(&hi[s], idx); return; }
    s = (s + 1) & HMASK;
  }
}

__global__ void k_markdup(const unsigned long long* __restrict__ keys,
                          const unsigned long long* __restrict__ hkey,
                          const int* __restrict__ hidx,
                          float* __restrict__ feat) {
  int i = blockIdx.x * blockDim.x + threadIdx.x;
  if (i >= BDIM * PDIM) return;
  int b = i / PDIM, idx = i % PDIM;
  unsigned long long key = keys[i];
  const unsigned long long* hk = hkey + (size_t)b * HSIZE;
  const int* hi = hidx + (size_t)b * HSIZE;
  unsigned int s = hash_slot(key);
  int keep = idx;
  for (int probe = 0; probe < HSIZE; ++probe) {
    if (hk[s] == key) { keep = hi[s]; break; }
    s = (s + 1) & HMASK;
  }
  if (keep != idx) {  // duplicate: poison row -> exp2(dot) == 0 for all centers
    float* f = feat + (size_t)i * 4;
    f[0] = 1e38f; f[1] = 0.0f; f[2] = 0.0f; f[3] = 0.0f;
  }
}

// ---------------------------------------------------------------- pass 1: column sums
// D = A(16x4) x B(4x16) via V_WMMA_F32_16X16X4_F32; props = exp2(D + const)
__global__ void __launch_bounds__(256)
k_colsum(const float* __restrict__ feat, const float* __restrict__ cvec,
         float* __restrict__ colsum) {
  int b  = blockIdx.x;        // batch
  int nt = blockIdx.y;        // 16-center tile
  int lane = threadIdx.x & 31, wid = threadIdx.x >> 5;
  int half = lane >> 4, l16 = lane & 15;
  int kb = half * 2;          // K-pair this lane holds (0,1) or (2,3)
  int col = nt * 16 + l16;

  const float* cv = cvec + ((size_t)b * KDIM + col) * 8;
  v2f Bm; Bm.x = cv[kb]; Bm.y = cv[kb + 1];
  float constk = cv[4];

  const float* fb = feat + (size_t)b * PDIM * 4;
  float csum = 0.0f;
  // scalar trip count (72), uniform per wave -> SALU loop, EXEC stays all-ones
#pragma unroll 1
  for (int it = 0; it < MT_COUNT / 8; ++it) {
    int row = (it * 8 + wid) * 16 + l16;
    const float* fr = fb + (size_t)row * 4 + kb;
    __builtin_prefetch(fb + (size_t)(row + 128) * 4 + kb, 0, 0);  // next tile
    v2f Am; Am.x = fr[0]; Am.y = fr[1];
    v8f acc = {};
    acc = __builtin_amdgcn_wmma_f32_16x16x4_f32(
        false, Am, false, Bm, (short)0, acc, false, false);
#pragma unroll
    for (int j = 0; j < 8; ++j)
      csum += __builtin_amdgcn_exp2f(acc[j] + constk);
  }
  csum += __shfl_xor(csum, 16, 32);              // combine M-halves of the column
  if (half == 0) atomicAdd(&colsum[(size_t)b * KDIM + col], csum);
}

// ---------------------------------------------------------------- weights
__global__ void k_weights(float* __restrict__ cvec, const float* __restrict__ colsum) {
  int t = blockIdx.x * blockDim.x + threadIdx.x;
  if (t >= BDIM * KDIM) return;
  cvec[(size_t)t * 8 + 6] = cvec[(size_t)t * 8 + 5] / colsum[t];
}

// ---------------------------------------------------------------- pass 2: output
__global__ void __launch_bounds__(256)
k_out(const float* __restrict__ feat, const float* __restrict__ cvec,
      const float* __restrict__ x, float* __restrict__ y) {
  int b    = blockIdx.x;                  // batch
  int mblk = blockIdx.y;                  // 72 blocks x 8 waves = 576 M-tiles
  int lane = threadIdx.x & 31, wid = threadIdx.x >> 5;
  int half = lane >> 4, l16 = lane & 15, kb = half * 2;
  int mt = mblk * 8 + wid;
  int row0 = mt * 16;

  const float* fb = feat + (size_t)b * PDIM * 4;
  int arow = row0 + l16;
  v2f Am; Am.x = fb[(size_t)arow * 4 + kb]; Am.y = fb[(size_t)arow * 4 + kb + 1];

  float racc[8];
#pragma unroll
  for (int j = 0; j < 8; ++j) racc[j] = 0.0f;

  const float* cb = cvec + (size_t)b * KDIM * 8;
#pragma unroll 1
  for (int nt = 0; nt < 16; ++nt) {
    const float* cv = cb + (size_t)(nt * 16 + l16) * 8;
    v2f Bm; Bm.x = cv[kb]; Bm.y = cv[kb + 1];
    float constk = cv[4], w = cv[6];
    v8f acc = {};
    acc = __builtin_amdgcn_wmma_f32_16x16x4_f32(
        false, Am, false, Bm, (short)0, acc, false, false);
#pragma unroll
    for (int j = 0; j < 8; ++j)
      racc[j] += __builtin_amdgcn_exp2f(acc[j] + constk) * w;
  }
  // reduce across the 16 columns held by this half-wave
#pragma unroll
  for (int j = 0; j < 8; ++j) {
    racc[j] += __shfl_xor(racc[j], 1, 32);
    racc[j] += __shfl_xor(racc[j], 2, 32);
    racc[j] += __shfl_xor(racc[j], 4, 32);
    racc[j] += __shfl_xor(racc[j], 8, 32);
  }
  if (l16 == 0) {                          // lanes 0 and 16 write rows 0-7 / 8-15
    const float* xb = x + (size_t)b * INSZ;
    float* yb = y + (size_t)b * OUTSZ;
#pragma unroll
    for (int j = 0; j < 8; ++j) {
      int row = row0 + j + half * 8;
      const float* fr = fb + (size_t)row * 4;
      if (fr[0] > 1e37f) continue;         // duplicate row -> zero contribution
      int t0 = (int)fr[1];
      int t1 = (int)fr[3];
      atomicAdd(&yb[t0], racc[j] * xb[t1]);
    }
  }
}

// ---------------------------------------------------------------- launch
extern "C" void kernel_launch(void* const* d_in, const int* in_sizes, int n_in,
                              void* d_out, int out_size, void* d_ws, size_t ws_size,
                              hipStream_t stream) {
  (void)in_sizes; (void)n_in; (void)out_size; (void)ws_size;
  const float* x    = (const float*)d_in[0];
  const float* res  = (const float*)d_in[1];
  const float* rr_u = (const float*)d_in[2];
  const float* g_u  = (const float*)d_in[3];
  float* y = (float*)d_out;

  char* ws = (char*)d_ws;
  size_t off = 0;
  auto carve = [&](size_t bytes) {
    void* p = ws + off;
    off = (off + bytes + 255) & ~(size_t)255;
    return p;
  };
  float*              feat   = (float*)carve((size_t)BDIM * PDIM * 4 * sizeof(float));
  unsigned long long* keys   = (unsigned long long*)carve((size_t)BDIM * PDIM * 8);
  float*              cvec   = (float*)carve((size_t)BDIM * KDIM * 8 * sizeof(float));
  float*              colsum = (float*)carve((size_t)BDIM * KDIM * sizeof(float));
  unsigned long long* hkey   = (unsigned long long*)carve((size_t)BDIM * HSIZE * 8);
  int*                hidx   = (int*)carve((size_t)BDIM * HSIZE * sizeof(int));

  int ninit = BDIM * HSIZE;
  k_init<<<(ninit + 255) / 256, 256, 0, stream>>>(hkey, hidx, colsum, y);

  int nbk = BDIM * KDIM;
  k_prep<<<(nbk + 255) / 256, 256, 0, stream>>>(res, rr_u, g_u, feat, keys, cvec);

  int npt = BDIM * PDIM;
  k_insert<<<(npt + 255) / 256, 256, 0, stream>>>(keys, hkey, hidx);
  k_markdup<<<(npt + 255) / 256, 256, 0, stream>>>(keys, hkey, hidx, feat);

  k_colsum<<<dim3(BDIM, KDIM / 16), 256, 0, stream>>>(feat, cvec, colsum);
  k_weights<<<(nbk + 255) / 256, 256, 0, stream>>>(cvec, colsum);
  k_out<<<dim3(BDIM, MT_COUNT / 8), 256, 0, stream>>>(feat, cvec, x, y);
}